// SampledSoftmaxXML_51986284151004
// MI455X (gfx1250) — compile-verified
//
#include <hip/hip_runtime.h>
#include <math.h>

// Problem constants (match reference)
#define B_DIM 256
#define L_LEN 128
#define K_POS 5
#define D_DIM 512
#define NNEG 2048
#define NUM_LABELS 670000
#define NORM_EPS 1e-4f

typedef __attribute__((ext_vector_type(2))) float v2f;
typedef __attribute__((ext_vector_type(8))) float v8f;

// ---------------- workspace layout (floats, 256B aligned) ----------------
// e          : B*D            = 131072 f  @ 0
// Wneg[d][n] : D*N            = 1048576 f @ 524288 B
// pos_logits : B*K            = 1280 f    @ 4718592 B
// neg_logits : B*N            = 524288 f  @ 4723712 B
// accum[2]   : numer, denom              @ 6820864 B
#define OFF_E       0
#define OFF_WNEG    (524288 / 4)
#define OFF_POS     (4718592 / 4)
#define OFF_NEG     (4723712 / 4)
#define OFF_ACC     (6820864 / 4)

// -------------------------------------------------------------------------
__global__ void init_accum(float* acc, float* out) {
    acc[0] = 0.0f;
    acc[1] = 0.0f;
    out[0] = 0.0f;
}

// One block per batch row; 512 threads = one per embedding dim.
__global__ __launch_bounds__(512) void embed_mean_norm(
    const int* __restrict__ indices, const float* __restrict__ mask,
    const float* __restrict__ table, float* __restrict__ e) {
    const int b = blockIdx.x;
    const int d = threadIdx.x;

    float acc = 0.0f, msum = 0.0f;
    for (int l = 0; l < L_LEN; ++l) {
        const float mk = mask[b * L_LEN + l];
        const int idx = indices[b * L_LEN + l];
        acc += mk * table[(long long)idx * D_DIM + d];
        msum += mk;
    }
    float val = acc / fmaxf(msum, 1.0f);

    __shared__ float red[512];
    red[d] = val * val;
    __syncthreads();
    for (int o = 256; o > 0; o >>= 1) {
        if (d < o) red[d] += red[d + o];
        __syncthreads();
    }
    const float nrm = sqrtf(red[0]);
    e[b * D_DIM + d] = val / fmaxf(nrm, NORM_EPS);
}

// Materialize Wneg[d][n] = W[d, neg_idx[n]]  (row-major [D][NNEG])
__global__ __launch_bounds__(256) void gather_wneg(
    const float* __restrict__ W, const int* __restrict__ negidx,
    float* __restrict__ Wneg) {
    const int i = blockIdx.x * 256 + threadIdx.x;   // 0 .. D*NNEG-1
    const int d = i / NNEG;
    const int n = i % NNEG;
    Wneg[i] = W[(long long)d * NUM_LABELS + negidx[n]];
}

// Per-(b,k) dot: pos[b,k] = e[b] . W[:, max(labels[b,k],0)]
__global__ __launch_bounds__(256) void pos_logits_kernel(
    const float* __restrict__ e, const float* __restrict__ W,
    const int* __restrict__ labels, float* __restrict__ pos) {
    const int bk = blockIdx.x;            // 0 .. B*K-1
    const int b = bk / K_POS;
    int col = labels[bk];
    if (col < 0) col = 0;

    float s = 0.0f;
    for (int d = threadIdx.x; d < D_DIM; d += 256)
        s += e[b * D_DIM + d] * W[(long long)d * NUM_LABELS + col];

    __shared__ float red[256];
    red[threadIdx.x] = s;
    __syncthreads();
    for (int o = 128; o > 0; o >>= 1) {
        if (threadIdx.x < o) red[threadIdx.x] += red[threadIdx.x + o];
        __syncthreads();
    }
    if (threadIdx.x == 0) pos[bk] = red[0];
}

// neg_logits[256,2048] = e[256,512] x Wneg[512,2048] with V_WMMA_F32_16X16X4_F32.
// Block = 256 threads (8 waves). Block tile: M=16, N=128 (16 per wave).
// K chunked at 64 through LDS.
#define KC 64
__global__ __launch_bounds__(256) void wmma_neg_gemm(
    const float* __restrict__ e, const float* __restrict__ Wneg,
    float* __restrict__ out) {
    __shared__ float Ash[16][KC + 4];      // +4 pad vs bank conflicts
    __shared__ float Bsh[8][KC][16];       // per-wave B tile [k][n]

    const int tid  = threadIdx.x;
    const int wave = tid >> 5;
    const int lane = tid & 31;
    const int half = lane >> 4;            // 0: lanes 0-15, 1: lanes 16-31
    const int lm   = lane & 15;
    const int mb   = blockIdx.x * 16;
    const int nb   = blockIdx.y * 128 + wave * 16;

    v8f c = {};
    for (int kc = 0; kc < D_DIM; kc += KC) {
        // --- A tile 16x64: one float4 per thread, cooperative across block
        {
            const int m  = tid >> 4;          // 0..15
            const int k4 = (tid & 15) * 4;    // 0..60
            const float4 a = *(const float4*)(e + (mb + m) * D_DIM + kc + k4);
            Ash[m][k4 + 0] = a.x; Ash[m][k4 + 1] = a.y;
            Ash[m][k4 + 2] = a.z; Ash[m][k4 + 3] = a.w;
        }
        // --- B tile 64x16 per wave: rows of Wneg are contiguous in n
        {
            const int fq = (lane & 3) * 4;    // n sub-offset 0,4,8,12
            const int kb = lane >> 2;         // 0..7
            #pragma unroll
            for (int j = 0; j < 8; ++j) {
                const int k = kb + j * 8;
                const float4 bv = *(const float4*)(Wneg + (kc + k) * NNEG + nb + fq);
                *(float4*)&Bsh[wave][k][fq] = bv;
            }
        }
        __syncthreads();

        // --- 16 WMMA steps over the 64-deep chunk (K=4 each)
        #pragma unroll
        for (int k0 = 0; k0 < KC; k0 += 4) {
            v2f a, b;
            // A-frag (16x4 f32): lane half selects K pair (ISA 7.12.2)
            a.x = Ash[lm][k0 + 2 * half + 0];
            a.y = Ash[lm][k0 + 2 * half + 1];
            // B-frag (4x16 f32): rows striped across lanes, same half split
            b.x = Bsh[wave][k0 + 2 * half + 0][lm];
            b.y = Bsh[wave][k0 + 2 * half + 1][lm];
            c = __builtin_amdgcn_wmma_f32_16x16x4_f32(
                    /*neg_a=*/false, a, /*neg_b=*/false, b,
                    /*c_mod=*/(short)0, c, /*reuse_a=*/false, /*reuse_b=*/false);
        }
        __syncthreads();
    }

    // C/D layout: VGPR r -> (M=r, N=lane) for lanes 0-15; (M=r+8, N=lane-16) for 16-31
    const int n = nb + lm;
    #pragma unroll
    for (int r = 0; r < 8; ++r) {
        const int m = mb + r + half * 8;
        out[m * NNEG + n] = c[r];
    }
}

// Per-row log-softmax over [pos(5) ++ neg(2048)], accumulate loss terms.
__global__ __launch_bounds__(256) void softmax_loss(
    const float* __restrict__ pos, const float* __restrict__ neg,
    const float* __restrict__ label_mask, float* __restrict__ accum) {
    const int b = blockIdx.x;
    const int t = threadIdx.x;
    __shared__ float red[256];

    float mx = -1e30f;
    for (int i = t; i < K_POS; i += 256) mx = fmaxf(mx, pos[b * K_POS + i]);
    for (int i = t; i < NNEG; i += 256)  mx = fmaxf(mx, neg[b * NNEG + i]);
    red[t] = mx;
    __syncthreads();
    for (int o = 128; o > 0; o >>= 1) {
        if (t < o) red[t] = fmaxf(red[t], red[t + o]);
        __syncthreads();
    }
    mx = red[0];
    __syncthreads();

    float se = 0.0f;
    for (int i = t; i < K_POS; i += 256) se += __expf(pos[b * K_POS + i] - mx);
    for (int i = t; i < NNEG; i += 256)  se += __expf(neg[b * NNEG + i] - mx);
    red[t] = se;
    __syncthreads();
    for (int o = 128; o > 0; o >>= 1) {
        if (t < o) red[t] += red[t + o];
        __syncthreads();
    }
    const float lse = logf(red[0]) + mx;

    if (t == 0) {
        float num = 0.0f, den = 0.0f;
        for (int k = 0; k < K_POS; ++k) {
            const float lm = label_mask[b * K_POS + k];
            num += lm * (pos[b * K_POS + k] - lse);
            den += lm;
        }
        atomicAdd(&accum[0], num);
        atomicAdd(&accum[1], den);
    }
}

__global__ void finalize(const float* __restrict__ accum, float* __restrict__ out) {
    out[0] = -accum[0] / (accum[1] + 1e-6f);
}

// -------------------------------------------------------------------------
extern "C" void kernel_launch(void* const* d_in, const int* in_sizes, int n_in,
                              void* d_out, int out_size, void* d_ws, size_t ws_size,
                              hipStream_t stream) {
    const int*   indices    = (const int*)  d_in[0];   // [B, L]
    const float* mask       = (const float*)d_in[1];   // [B, L]
    const int*   labels     = (const int*)  d_in[2];   // [B, K]
    const float* label_mask = (const float*)d_in[3];   // [B, K]
    const int*   neg_idx    = (const int*)  d_in[4];   // [N]
    const float* table      = (const float*)d_in[5];   // [NUM_FEATURES, D]
    const float* W          = (const float*)d_in[6];   // [D, NUM_LABELS]
    float* out = (float*)d_out;

    float* ws   = (float*)d_ws;
    float* e    = ws + OFF_E;
    float* Wneg = ws + OFF_WNEG;
    float* pos  = ws + OFF_POS;
    float* neg  = ws + OFF_NEG;
    float* acc  = ws + OFF_ACC;

    init_accum<<<1, 1, 0, stream>>>(acc, out);
    embed_mean_norm<<<B_DIM, 512, 0, stream>>>(indices, mask, table, e);
    gather_wneg<<<(D_DIM * NNEG) / 256, 256, 0, stream>>>(W, neg_idx, Wneg);
    pos_logits_kernel<<<B_DIM * K_POS, 256, 0, stream>>>(e, W, labels, pos);
    wmma_neg_gemm<<<dim3(B_DIM / 16, NNEG / 128), 256, 0, stream>>>(e, Wneg, neg);
    softmax_loss<<<B_DIM, 256, 0, stream>>>(pos, neg, label_mask, acc);
    finalize<<<1, 1, 0, stream>>>(acc, out);
}